// AlignUniform_34162169872914
// MI455X (gfx1250) — compile-verified
//
#include <hip/hip_runtime.h>
#include <math.h>

// AlignUniform loss for MI455X (gfx1250, wave32).
//
// Work: 2x Gram matrices (8192 x 8192 x 128 fp32) -> exp -> global sum.
// Compute-bound (34 GFLOP vs 8 MB traffic); precision-sensitive (exp/log of
// sums), so use V_WMMA_F32_16X16X4_F32 (full fp32 matrix pipe) rather than
// bf16 WMMA. Accumulate global sums in fp64.
//
// Gram kernel: each wave owns a 32x32 output block (2x2 WMMA tiles) so each
// k-step does 4 loads -> 4 WMMAs (2x operand reuse vs 1x1 tiling), and only
// jt >= it tile-pairs are computed (symmetry; strictly-upper tiles weighted
// 2x, diagonal tiles 1x; upper-triangle sum recovered as (S_full - N)/2).

typedef float v2f __attribute__((ext_vector_type(2)));
typedef float v8f __attribute__((ext_vector_type(8)));

#define DIMD 128  // feature dimension (fixed by reference)

// ---------------------------------------------------------------- init ------
__global__ __launch_bounds__(32) void au_init(double* __restrict__ acc) {
    if (threadIdx.x < 3) acc[threadIdx.x] = 0.0;
}

// ------------------------------------------------- normalize + align dot ----
// One wave per row: lane holds 4 consecutive floats of the 128-dim row.
__global__ __launch_bounds__(256) void au_normalize(
    const float* __restrict__ q, const float* __restrict__ k,
    float* __restrict__ qn, float* __restrict__ kn,
    double* __restrict__ acc) {
    const int wave = threadIdx.x >> 5;
    const int lane = threadIdx.x & 31;
    const int row  = blockIdx.x * 8 + wave;
    const int base = row * DIMD + lane * 4;

    float4 qv = *(const float4*)(q + base);
    float4 kv = *(const float4*)(k + base);

    float sq = qv.x * qv.x + qv.y * qv.y + qv.z * qv.z + qv.w * qv.w;
    float sk = kv.x * kv.x + kv.y * kv.y + kv.z * kv.z + kv.w * kv.w;
    #pragma unroll
    for (int off = 16; off >= 1; off >>= 1) {
        sq += __shfl_xor(sq, off, 32);
        sk += __shfl_xor(sk, off, 32);
    }
    const float invq = 1.0f / fmaxf(sqrtf(sq), 1e-12f);
    const float invk = 1.0f / fmaxf(sqrtf(sk), 1e-12f);
    qv.x *= invq; qv.y *= invq; qv.z *= invq; qv.w *= invq;
    kv.x *= invk; kv.y *= invk; kv.z *= invk; kv.w *= invk;
    *(float4*)(qn + base) = qv;
    *(float4*)(kn + base) = kv;

    float d = qv.x * kv.x + qv.y * kv.y + qv.z * kv.z + qv.w * kv.w;
    #pragma unroll
    for (int off = 16; off >= 1; off >>= 1) d += __shfl_xor(d, off, 32);
    if (lane == 0) atomicAdd(&acc[0], (double)d);
}

// ------------------------------------------- Gram tiles + exp + reduction ---
// fp32 A(16x4) and B(4x16) fragments have identical per-lane layouts for the
// Gram case: lane L -> two consecutive floats of row base+(L&15), at column
// k0 + 2*(L>>4).  (ISA 7.12.2: A 16x4 table; B "row striped across lanes".)
__global__ __launch_bounds__(256) void au_gram_expsum(
    const float* __restrict__ qn, const float* __restrict__ kn,
    double* __restrict__ acc, int N) {
    const float* __restrict__ X = (blockIdx.z == 0) ? qn : kn;
    const int wave = threadIdx.x >> 5;
    const int lane = threadIdx.x & 31;

    const int it32 = blockIdx.y * 8 + wave;  // 32-row block index
    const int jt32 = blockIdx.x;             // 32-col block index

    float local = 0.0f;
    if (it32 <= jt32) {  // wave-uniform: EXEC stays all-ones for WMMA
        const int row_base = it32 * 32;
        const int col_base = jt32 * 32;
        const int r  = lane & 15;
        const int kh = (lane >> 4) << 1;  // 0 or 2

        const float* __restrict__ a0 = X + (size_t)(row_base + r) * DIMD + kh;
        const float* __restrict__ a1 = a0 + 16 * DIMD;
        const float* __restrict__ b0 = X + (size_t)(col_base + r) * DIMD + kh;
        const float* __restrict__ b1 = b0 + 16 * DIMD;

        v8f c00 = {}, c01 = {}, c10 = {}, c11 = {};
        #pragma unroll 4
        for (int k0 = 0; k0 < DIMD; k0 += 4) {
            v2f av0 = *(const v2f*)(a0 + k0);
            v2f av1 = *(const v2f*)(a1 + k0);
            v2f bv0 = *(const v2f*)(b0 + k0);
            v2f bv1 = *(const v2f*)(b1 + k0);
            // 8 args: (neg_a, A, neg_b, B, c_mod, C, reuse_a, reuse_b)
            c00 = __builtin_amdgcn_wmma_f32_16x16x4_f32(
                false, av0, false, bv0, (short)0, c00, false, false);
            c01 = __builtin_amdgcn_wmma_f32_16x16x4_f32(
                false, av0, false, bv1, (short)0, c01, false, false);
            c10 = __builtin_amdgcn_wmma_f32_16x16x4_f32(
                false, av1, false, bv0, (short)0, c10, false, false);
            c11 = __builtin_amdgcn_wmma_f32_16x16x4_f32(
                false, av1, false, bv1, (short)0, c11, false, false);
        }

        // exp(-2 * max(2 - 2s, 0)) == exp(min(-4 + 4s, 0))
        float t = 0.0f;
        #pragma unroll
        for (int v = 0; v < 8; ++v) {
            t += __expf(fminf(-4.0f + 4.0f * c00[v], 0.0f));
            t += __expf(fminf(-4.0f + 4.0f * c01[v], 0.0f));
            t += __expf(fminf(-4.0f + 4.0f * c10[v], 0.0f));
            t += __expf(fminf(-4.0f + 4.0f * c11[v], 0.0f));
        }
        // strictly-upper blocks count twice (symmetry), diagonal blocks once
        local = (it32 == jt32) ? t : 2.0f * t;
    }

    #pragma unroll
    for (int off = 16; off >= 1; off >>= 1) local += __shfl_xor(local, off, 32);

    __shared__ float partial[8];
    if (lane == 0) partial[wave] = local;
    __syncthreads();
    if (threadIdx.x == 0) {
        float t = 0.0f;
        #pragma unroll
        for (int w = 0; w < 8; ++w) t += partial[w];
        atomicAdd(&acc[1 + blockIdx.z], (double)t);
    }
}

// ------------------------------------------------------------- finalize -----
__global__ __launch_bounds__(32) void au_finalize(
    const double* __restrict__ acc, float* __restrict__ out, int N) {
    if (threadIdx.x != 0 || blockIdx.x != 0) return;
    const double dot   = acc[0];
    const double sum_q = acc[1];  // full-matrix sum incl. diagonal (~N)
    const double sum_k = acc[2];
    const double align  = 2.0 - 2.0 * dot / (double)N;
    const double npairs = 0.5 * (double)N * (double)(N - 1);
    const double uq = log(((sum_q - (double)N) * 0.5) / npairs);
    const double uk = log(((sum_k - (double)N) * 0.5) / npairs);
    out[0] = (float)(align + 0.5 * (uq + uk));
}

// ---------------------------------------------------------------- launch ----
extern "C" void kernel_launch(void* const* d_in, const int* in_sizes, int n_in,
                              void* d_out, int out_size, void* d_ws, size_t ws_size,
                              hipStream_t stream) {
    const float* q = (const float*)d_in[0];
    const float* k = (const float*)d_in[1];
    const int N = in_sizes[0] / DIMD;  // 8192

    float*  qn  = (float*)d_ws;
    float*  kn  = qn + (size_t)N * DIMD;
    double* acc = (double*)(kn + (size_t)N * DIMD);  // 8 MB offset: aligned
    float*  out = (float*)d_out;

    au_init<<<1, 32, 0, stream>>>(acc);
    au_normalize<<<N / 8, 256, 0, stream>>>(q, k, qn, kn, acc);
    // 32x32 blocks per wave; 8 waves per block => grid (N/32, N/256, 2)
    au_gram_expsum<<<dim3(N / 32, N / 256, 2), 256, 0, stream>>>(qn, kn, acc, N);
    au_finalize<<<1, 32, 0, stream>>>(acc, out, N);
}